// CSGNet_76235669504707
// MI455X (gfx1250) — compile-verified
//
#include <hip/hip_runtime.h>
#include <cstdint>
#include <cstddef>

// ---------------- problem constants (match reference) ----------------
#define BB   256          // batch (graphs)
#define NN   2207         // nodes per graph
#define MM   16           // mid channels
#define EE   16777216     // edges
#define CC1  12
#define CC2  4
#define HH1  256
#define HH2  64
#define BNODES (BB * NN)  // 564992

static __device__ __constant__ float kEps = 1e-5f;
#define BN_SCALE 0.9999950000374997f   // 1/sqrt(1+1e-5)

// ---------------- vector / WMMA fragment types ----------------
typedef __attribute__((ext_vector_type(2))) float v2f;
typedef __attribute__((ext_vector_type(8))) float v8f;
typedef __attribute__((ext_vector_type(4))) float f32x4;
typedef __attribute__((ext_vector_type(4))) int   i32x4;

// ---------------- gfx1250 async global->LDS staging ----------------
#if __has_builtin(__builtin_amdgcn_global_load_async_to_lds_b128)
#define HAVE_ASYNC_LDS 1
__device__ __forceinline__ void lds_copy16(const float* g, float* l) {
    // builtin signature (from hipcc diagnostic): (int4* src, int4* lds, imm offset, imm cpol)
    __builtin_amdgcn_global_load_async_to_lds_b128((i32x4*)(g), (i32x4*)(l), 0, 0);
}
#if __has_builtin(__builtin_amdgcn_s_wait_asynccnt)
__device__ __forceinline__ void async_wait0() { __builtin_amdgcn_s_wait_asynccnt(0); }
#else
__device__ __forceinline__ void async_wait0() { asm volatile("s_wait_asynccnt 0" ::: "memory"); }
#endif
#else
#define HAVE_ASYNC_LDS 0
__device__ __forceinline__ void lds_copy16(const float* g, float* l) {
    *(f32x4*)l = *(const f32x4*)g;
}
__device__ __forceinline__ void async_wait0() {}
#endif

// ---------------- kernel 0: zero the aggregation buffer ----------------
__global__ void k_zero(float* __restrict__ p, int n) {
    int i = blockIdx.x * blockDim.x + threadIdx.x;
    if (i < n) p[i] = 0.0f;
}

// ---------------- kernel 1: edge scatter-add (bandwidth bound) ----------------
// agg[dst] += x[src] * w  over 16.7M edges. Edge streams are touched once ->
// non-temporal 128-bit loads. x/agg are L2 resident (2.2MB each).
__global__ void k_edge_scatter(const int* __restrict__ src,
                               const int* __restrict__ dst,
                               const float* __restrict__ w,
                               const float* __restrict__ x,
                               float* __restrict__ agg) {
    size_t t = (size_t)blockIdx.x * blockDim.x + threadIdx.x;
    size_t i = t * 4;
    i32x4 s  = __builtin_nontemporal_load((const i32x4*)(src + i));
    i32x4 d  = __builtin_nontemporal_load((const i32x4*)(dst + i));
    f32x4 wv = __builtin_nontemporal_load((const f32x4*)(w + i));
    unsafeAtomicAdd(&agg[d.x], x[s.x] * wv.x);
    unsafeAtomicAdd(&agg[d.y], x[s.y] * wv.y);
    unsafeAtomicAdd(&agg[d.z], x[s.z] * wv.z);
    unsafeAtomicAdd(&agg[d.w], x[s.w] * wv.w);
}

// ---------------- kernel 2: GraphConv elementwise + per-batch LayerNorm ----
__global__ void k_h_layernorm(const float* __restrict__ agg,
                              const float* __restrict__ x,
                              const float* __restrict__ w_root,
                              const float* __restrict__ w_rel,
                              const float* __restrict__ b_rel,
                              const float* __restrict__ ln_g,
                              const float* __restrict__ ln_b,
                              float* __restrict__ h_t) {
    __shared__ float rs[256];
    __shared__ float rq[256];
    __shared__ float stats[2];
    const int b = blockIdx.x, tid = threadIdx.x;

    float wr[MM], wo[MM], br[MM];
#pragma unroll
    for (int m = 0; m < MM; ++m) { wr[m] = w_rel[m]; wo[m] = w_root[m]; br[m] = b_rel[m]; }

    float s = 0.0f, ss = 0.0f;
    for (int n = tid; n < NN; n += 256) {
        const float a  = agg[b * NN + n];
        const float xv = x[b * NN + n];
#pragma unroll
        for (int m = 0; m < MM; ++m) {
            float v = fmaf(a, wr[m], fmaf(xv, wo[m], br[m]));
            v = fmaxf(v, 0.0f);
            s += v; ss += v * v;
        }
    }
    rs[tid] = s; rq[tid] = ss;
    __syncthreads();
    for (int off = 128; off > 0; off >>= 1) {
        if (tid < off) { rs[tid] += rs[tid + off]; rq[tid] += rq[tid + off]; }
        __syncthreads();
    }
    if (tid == 0) {
        const float inv = 1.0f / (float)(NN * MM);
        const float mu  = rs[0] * inv;
        const float var = rq[0] * inv - mu * mu;   // biased, matches torch LN
        stats[0] = mu;
        stats[1] = rsqrtf(var + kEps);
    }
    __syncthreads();
    const float mu = stats[0], rstd = stats[1];

    for (int n = tid; n < NN; n += 256) {
        const float a  = agg[b * NN + n];
        const float xv = x[b * NN + n];
#pragma unroll
        for (int m = 0; m < MM; ++m) {
            float v = fmaxf(fmaf(a, wr[m], fmaf(xv, wo[m], br[m])), 0.0f);
            v = (v - mu) * rstd * ln_g[n * MM + m] + ln_b[n * MM + m];
            h_t[(size_t)b * MM * NN + (size_t)m * NN + n] = v;
        }
    }
}

// ---------------- kernel 3: fused 1x1 conv stack (16->12->4) ----------------
__global__ void k_conv_stack(const float* __restrict__ h_t,
                             const float* __restrict__ gc1_w, const float* __restrict__ gc1_b,
                             const float* __restrict__ bn1_g, const float* __restrict__ bn1_b,
                             const float* __restrict__ gc2_w, const float* __restrict__ gc2_b,
                             const float* __restrict__ bn2_g, const float* __restrict__ bn2_b,
                             float* __restrict__ z) {
    const int idx = blockIdx.x * blockDim.x + threadIdx.x;
    if (idx >= BNODES) return;
    const int b = idx / NN, n = idx % NN;

    float hv[MM];
#pragma unroll
    for (int m = 0; m < MM; ++m)
        hv[m] = h_t[(size_t)b * MM * NN + (size_t)m * NN + n];

    float y1[CC1];
#pragma unroll
    for (int o = 0; o < CC1; ++o) {
        float acc = gc1_b[o];
#pragma unroll
        for (int m = 0; m < MM; ++m) acc = fmaf(gc1_w[o * MM + m], hv[m], acc);
        y1[o] = fmaxf(acc, 0.0f) * (BN_SCALE * bn1_g[o]) + bn1_b[o];
    }
#pragma unroll
    for (int o = 0; o < CC2; ++o) {
        float acc = gc2_b[o];
#pragma unroll
        for (int c = 0; c < CC1; ++c) acc = fmaf(gc2_w[o * CC1 + c], y1[c], acc);
        float v = fmaxf(acc, 0.0f) * (BN_SCALE * bn2_g[o]) + bn2_b[o];
        z[(size_t)b * (CC2 * NN) + (size_t)o * NN + n] = v;
    }
}

// ---------------- kernels 4/5: LDS-tiled fp32 WMMA GEMM ---------------------
// D = relu(((A[M x K] @ W[K x N]) + bias) * (BN_SCALE*g) + bb)
// 256 threads = 8 waves in a 2x4 grid of 16x16 tiles -> 32x64 WG tile.
// K blocked by 32; A/B panels staged to LDS with GLOBAL_LOAD_ASYNC_TO_LDS_B128
// (ASYNCcnt) when available. Padded strides (A:36, B:80 floats) keep the
// per-k fragment reads bank-conflict-free for both 16-lane halves.
#define KB     32
#define A_PAD  36
#define B_PAD  80

__global__ void k_wmma_fc(const float* __restrict__ A,
                          const float* __restrict__ W,
                          const float* __restrict__ bias,
                          const float* __restrict__ g,
                          const float* __restrict__ bb,
                          float* __restrict__ D,
                          int K, int Ndim) {
    __shared__ __align__(16) float As[KB * A_PAD];   // As[r*A_PAD + k], r=0..31
    __shared__ __align__(16) float Bs[KB * B_PAD];   // Bs[k*B_PAD + c], c=0..63

    const int nBlocksN = Ndim >> 6;                  // 64-wide column blocks
    const int blockRow = blockIdx.x / nBlocksN;
    const int blockCol = blockIdx.x % nBlocksN;
    const int tid  = threadIdx.x;
    const int lane = tid & 31;
    const int wid  = tid >> 5;                       // 0..7
    const int waveRow = wid >> 2;                    // 0..1
    const int waveCol = wid & 3;                     // 0..3
    const int half = lane >> 4;                      // frag K-half select
    const int l15  = lane & 15;

    // staging assignments (chunk = 16B)
    const int ar = tid >> 3;                         // A: row 0..31
    const int ak = (tid & 7) << 2;                   //    k-offset 0,4,..,28
    const int br0 = tid >> 4;                        // B chunk 0: row 0..15
    const int bc0 = (tid & 15) << 2;                 //    col-offset
    const int br1 = br0 + 16;                        // B chunk 1: row 16..31

    const float* Arow = A + (size_t)(blockRow * 32 + ar) * K + ak;
    const float* Brow0 = W + (size_t)br0 * Ndim + blockCol * 64 + bc0;
    const float* Brow1 = W + (size_t)br1 * Ndim + blockCol * 64 + bc0;

    v8f acc = {};

    for (int k0 = 0; k0 < K; k0 += KB) {
        const int kb = (K - k0 < KB) ? (K - k0) : KB;

        // ---- stage A (32 x kb) and B (kb x 64) into LDS ----
        if (ak < kb)
            lds_copy16(Arow + k0, &As[ar * A_PAD + ak]);
        if (br0 < kb)
            lds_copy16(Brow0 + (size_t)k0 * Ndim, &Bs[br0 * B_PAD + bc0]);
        if (br1 < kb)
            lds_copy16(Brow1 + (size_t)k0 * Ndim, &Bs[br1 * B_PAD + bc0]);
        // prefetch next B panel into GL2 (streamed once per column block)
        if (k0 + KB < K)
            __builtin_prefetch(Brow0 + (size_t)(k0 + KB) * Ndim, 0, 0);

        async_wait0();
        __syncthreads();

        // ---- compute: kb/4 WMMA steps per wave ----
        const float* aRowBase = &As[(waveRow * 16 + l15) * A_PAD + 2 * half];
        const float* bColBase = &Bs[(2 * half) * B_PAD + waveCol * 16 + l15];
        for (int kk = 0; kk < kb; kk += 4) {
            v2f a;  a.x  = aRowBase[kk];
                    a.y  = aRowBase[kk + 1];
            v2f bf; bf.x = bColBase[kk * B_PAD];
                    bf.y = bColBase[(kk + 1) * B_PAD];
            acc = __builtin_amdgcn_wmma_f32_16x16x4_f32(
                false, a, false, bf, (short)0, acc, false, false);
        }
        __syncthreads();   // protect LDS before next staging round
    }

    // ---- epilogue: bias + eval-BN + relu ----
    const int rowBase = blockRow * 32 + waveRow * 16;
    const int col     = blockCol * 64 + waveCol * 16 + l15;
    const float sc = BN_SCALE * g[col];
    const float bo = bb[col];
    const float bi = bias[col];
#pragma unroll
    for (int r = 0; r < 8; ++r) {
        const int row = rowBase + r + 8 * half;   // 16x16 f32 C/D VGPR layout
        float v = (acc[r] + bi) * sc + bo;
        D[(size_t)row * Ndim + col] = fmaxf(v, 0.0f);
    }
}

// ---------------- kernel 6: final [64 -> 1] layer ----------------
__global__ void k_fc_final(const float* __restrict__ z2,
                           const float* __restrict__ w,
                           const float* __restrict__ b,
                           float* __restrict__ out) {
    const int i = blockIdx.x * blockDim.x + threadIdx.x;
    if (i >= BB) return;
    float acc = b[0];
#pragma unroll
    for (int j = 0; j < HH2; ++j) acc = fmaf(z2[i * HH2 + j], w[j], acc);
    out[i] = acc;
}

// ---------------- launcher ----------------
extern "C" void kernel_launch(void* const* d_in, const int* in_sizes, int n_in,
                              void* d_out, int out_size, void* d_ws, size_t ws_size,
                              hipStream_t stream) {
    const float* x       = (const float*)d_in[0];
    const int*   eidx    = (const int*)  d_in[1];
    const float* ew      = (const float*)d_in[2];
    const float* w_root  = (const float*)d_in[3];
    const float* w_rel   = (const float*)d_in[4];
    const float* b_rel   = (const float*)d_in[5];
    const float* ln_g    = (const float*)d_in[6];
    const float* ln_b    = (const float*)d_in[7];
    const float* gc1_w   = (const float*)d_in[8];
    const float* gc1_b   = (const float*)d_in[9];
    const float* bn1_g   = (const float*)d_in[10];
    const float* bn1_b   = (const float*)d_in[11];
    const float* gc2_w   = (const float*)d_in[12];
    const float* gc2_b   = (const float*)d_in[13];
    const float* bn2_g   = (const float*)d_in[14];
    const float* bn2_b   = (const float*)d_in[15];
    const float* fc_w1   = (const float*)d_in[16];
    const float* fc_b1   = (const float*)d_in[17];
    const float* fbn1_g  = (const float*)d_in[18];
    const float* fbn1_b  = (const float*)d_in[19];
    const float* fc_w2   = (const float*)d_in[20];
    const float* fc_b2   = (const float*)d_in[21];
    const float* fbn2_g  = (const float*)d_in[22];
    const float* fbn2_b  = (const float*)d_in[23];
    const float* fc1_w   = (const float*)d_in[24];
    const float* fc1_b   = (const float*)d_in[25];
    float*       out     = (float*)d_out;

    // workspace layout (floats), 256-float aligned blocks, ~46 MB total
    float* ws  = (float*)d_ws;
    float* agg = ws;                              // 564,992
    float* h_t = ws + 565248;                     // 9,039,872  [B][M][N]
    float* z   = ws + 565248 + 9039872;           // 2,259,968  [B][C2*N]
    float* z1  = ws + 11865600;                   // 65,536     [B][H1]
    float* z2  = ws + 11931392;                   // 16,384     [B][H2]

    // 0) clear agg
    k_zero<<<(BNODES + 255) / 256, 256, 0, stream>>>(agg, BNODES);

    // 1) edge scatter: 4 edges/thread
    k_edge_scatter<<<EE / (256 * 4), 256, 0, stream>>>(
        eidx, eidx + EE, ew, x, agg);

    // 2) GraphConv elementwise + LayerNorm, one block per batch
    k_h_layernorm<<<BB, 256, 0, stream>>>(
        agg, x, w_root, w_rel, b_rel, ln_g, ln_b, h_t);

    // 3) conv stacks (exactly BNODES = 2207*256 threads)
    k_conv_stack<<<BNODES / 256, 256, 0, stream>>>(
        h_t, gc1_w, gc1_b, bn1_g, bn1_b, gc2_w, gc2_b, bn2_g, bn2_b, z);

    // 4) FC1: [256 x 8828] @ [8828 x 256] -> z1; 8x4 WGs of 32x64 tiles
    k_wmma_fc<<<(BB / 32) * (HH1 / 64), 256, 0, stream>>>(
        z, fc_w1, fc_b1, fbn1_g, fbn1_b, z1, CC2 * NN, HH1);

    // 5) FC2: [256 x 256] @ [256 x 64] -> z2; 8x1 WGs
    k_wmma_fc<<<(BB / 32) * (HH2 / 64), 256, 0, stream>>>(
        z1, fc_w2, fc_b2, fbn2_g, fbn2_b, z2, HH1, HH2);

    // 6) final dot: [B,64] @ [64,1] + b
    k_fc_final<<<1, BB, 0, stream>>>(z2, fc1_w, fc1_b, out);
}